// MyConvLayer_28638841930103
// MI455X (gfx1250) — compile-verified
//
#include <hip/hip_runtime.h>

typedef float v2f __attribute__((ext_vector_type(2)));
typedef float v8f __attribute__((ext_vector_type(8)));

#define HW_DIM   128
#define F_OUT    256
#define KDIM     2304   // C*3*3 = 256*9
#define NROI     8192
#define HOWO     16384  // 128*128
#define KTILE    16
#define LDA      20     // padded stride for A (128 x 16 tile), 16B aligned rows
#define LDB      136    // padded stride for B (16 x 128 tile), conflict-free frag reads

// ---------------------------------------------------------------------------
// Kernel 1: fill every output element with its bias value (float4 stores).
// out is (1, 256, 128, 128) f32; each f covers 16384 contiguous floats.
// ---------------------------------------------------------------------------
__global__ __launch_bounds__(256) void bias_fill_kernel(
    const float* __restrict__ bias, float* __restrict__ out) {
  const int g = blockIdx.x * 256 + threadIdx.x;   // index over float4s
  const int f = g >> 12;                           // 16384/4 = 4096 float4 per f
  const float bv = bias[f];
  reinterpret_cast<float4*>(out)[g] = make_float4(bv, bv, bv, bv);
}

// ---------------------------------------------------------------------------
// Kernel 2: ROI im2col GEMM via V_WMMA_F32_16X16X4_F32.
// D[256, 8192] = W[256, 2304] x Xg[2304, 8192], scattered into out via roi.
// Block tile 128(M) x 128(N), 8 waves (2x4), 8 WMMA tiles per wave.
// ---------------------------------------------------------------------------
__global__ __launch_bounds__(256) void conv_roi_wmma_kernel(
    const float* __restrict__ X,     // (256, 128, 128)
    const float* __restrict__ W,     // (256, 2304) row-major
    const float* __restrict__ bias,  // (256)
    const int*   __restrict__ roi,   // (8192), sorted
    float* __restrict__ out) {       // (256, 16384)
  __shared__ float As[128 * LDA];    // 10240 B
  __shared__ float Bs[KTILE * LDB];  //  8704 B

  const int tid   = threadIdx.x;
  const int lane  = tid & 31;
  const int wave  = tid >> 5;
  const int waveM = wave >> 2;       // 0..1 -> 64 rows each
  const int waveN = wave & 3;        // 0..3 -> 32 cols each
  const int lrow  = lane & 15;
  const int lhi   = lane >> 4;       // 0/1 (K pair / M-high selector)

  const int m0 = blockIdx.y * 128;
  const int n0 = blockIdx.x * 128;

  // --- B gather mapping: thread owns column tn, handles 8 k-rows per tile ---
  const int tn   = tid & 127;
  const int kgrp = tid >> 7;                 // 0/1 -> kk 0..7 / 8..15
  const int p_my = roi[n0 + tn];
  const int ry   = p_my >> 7;                // output row
  const int rx   = p_my & 127;               // output col

  // --- A load mapping: thread loads float4; 2 rows per thread ---
  const int arow = tid >> 2;                 // 0..63
  const int aq   = tid & 3;                  // quad within 16-wide row

  v8f acc[4][2];
  const v8f vzero = {};
#pragma unroll
  for (int mt = 0; mt < 4; ++mt)
#pragma unroll
    for (int nt = 0; nt < 2; ++nt) acc[mt][nt] = vzero;

  for (int k0 = 0; k0 < KDIM; k0 += KTILE) {
    // ---- stage A tile: W[m0..m0+127, k0..k0+15] ----
#pragma unroll
    for (int rep = 0; rep < 2; ++rep) {
      const int row = arow + rep * 64;
      const float4 w4 =
          *reinterpret_cast<const float4*>(W + (m0 + row) * KDIM + k0 + aq * 4);
      *reinterpret_cast<float4*>(&As[row * LDA + aq * 4]) = w4;
    }
    // ---- stage B tile: gathered im2col Xg[k0..k0+15, n0..n0+127] ----
#pragma unroll
    for (int u = 0; u < 8; ++u) {
      const int kk  = kgrp * 8 + u;
      const int k   = k0 + kk;
      const int ch  = k / 9;
      const int rem = k - ch * 9;
      const int fi  = rem / 3;
      const int fj  = rem - fi * 3;
      const int y   = ry + fi - 1;           // PAD = 1
      const int x   = rx + fj - 1;
      float v = 0.0f;
      if ((unsigned)y < 128u && (unsigned)x < 128u)
        v = X[(ch << 14) + (y << 7) + x];
      Bs[kk * LDB + tn] = v;
    }
    if (k0 + KTILE < KDIM)  // hint next A tile into cache (global_prefetch_b8)
      __builtin_prefetch(W + (m0 + arow) * KDIM + k0 + KTILE, 0, 0);
    __syncthreads();

    // ---- 4 K-substeps of 4, 8 WMMA tiles each: 32 v_wmma per K-tile ----
#pragma unroll
    for (int ks = 0; ks < 4; ++ks) {
      const int kb = ks * 4;
      // A frags: A[m][k]: lane = m + 16*(k/2), vgpr = k%2 -> contiguous pair
      v2f afrag[4];
#pragma unroll
      for (int mt = 0; mt < 4; ++mt) {
        const int m = waveM * 64 + mt * 16 + lrow;
        afrag[mt] = *reinterpret_cast<const v2f*>(&As[m * LDA + kb + lhi * 2]);
      }
      // B frags: B[k][n]: lane = n + 16*(k/2), vgpr = k%2 -> row-strided pair
      v2f bfrag[2];
#pragma unroll
      for (int nt = 0; nt < 2; ++nt) {
        const int n  = waveN * 32 + nt * 16 + lrow;
        const int kk = kb + lhi * 2;
        v2f b;
        b.x = Bs[kk * LDB + n];
        b.y = Bs[(kk + 1) * LDB + n];
        bfrag[nt] = b;
      }
#pragma unroll
      for (int mt = 0; mt < 4; ++mt)
#pragma unroll
        for (int nt = 0; nt < 2; ++nt)
          acc[mt][nt] = __builtin_amdgcn_wmma_f32_16x16x4_f32(
              false, afrag[mt], false, bfrag[nt],
              (short)0, acc[mt][nt], false, false);
    }
    __syncthreads();
  }

  // ---- epilogue: scatter acc + bias into out[f*16384 + roi[n]] ----
  // C/D layout: VGPR v, lanes 0-15 -> M=v, N=lane; lanes 16-31 -> M=v+8.
#pragma unroll
  for (int mt = 0; mt < 4; ++mt) {
    const int fbase = m0 + waveM * 64 + mt * 16 + lhi * 8;
#pragma unroll
    for (int nt = 0; nt < 2; ++nt) {
      const int nn = n0 + waveN * 32 + nt * 16 + lrow;
      const int p  = roi[nn];
#pragma unroll
      for (int v = 0; v < 8; ++v) {
        const int f = fbase + v;
        out[(f << 14) + p] = acc[mt][nt][v] + bias[f];
      }
    }
  }
}

// ---------------------------------------------------------------------------
extern "C" void kernel_launch(void* const* d_in, const int* in_sizes, int n_in,
                              void* d_out, int out_size, void* d_ws, size_t ws_size,
                              hipStream_t stream) {
  const float* X    = (const float*)d_in[0];   // (1,256,128,128) f32
  const float* W    = (const float*)d_in[1];   // (256,256,3,3)   f32
  const float* B    = (const float*)d_in[2];   // (256)           f32
  const int*   roi  = (const int*)d_in[3];     // (8192)          i32
  float*       out  = (float*)d_out;           // (1,256,128,128) f32

  // 1) fill entire output with bias
  bias_fill_kernel<<<dim3((F_OUT * HOWO / 4) / 256), 256, 0, stream>>>(B, out);
  // 2) overwrite ROI columns with GEMM result + bias
  conv_roi_wmma_kernel<<<dim3(NROI / 128, F_OUT / 128), 256, 0, stream>>>(
      X, W, B, roi, out);
}